// TGATHead_82549271429769
// MI455X (gfx1250) — compile-verified
//
#include <hip/hip_runtime.h>
#include <hip/hip_bf16.h>

typedef __attribute__((ext_vector_type(2))) float v2f;
typedef __attribute__((ext_vector_type(8))) float v8f;

// -------- float atomic max via sign-aware integer atomics --------
__device__ __forceinline__ void atomicMaxF32(float* addr, float val) {
    if (val >= 0.0f)
        atomicMax((int*)addr, __float_as_int(val));
    else
        atomicMin((unsigned int*)addr, (unsigned int)__float_as_int(val));
}

// 1) zero out / init max & den
__global__ void tgat_init(float* __restrict__ out, float* __restrict__ mx,
                          float* __restrict__ den, int N, int total) {
    int i = blockIdx.x * blockDim.x + threadIdx.x;
    if (i < total) out[i] = 0.0f;
    if (i < N) { mx[i] = -__builtin_inff(); den[i] = 0.0f; }
}

// 2) fold attention vector a into W / Wt:  wa1 = W^T a[0:64], wa2 = W^T a[64:128], wta = Wt^T a[128:192]
__global__ void tgat_fold_a(const float* __restrict__ W, const float* __restrict__ Wt,
                            const float* __restrict__ a,
                            float* __restrict__ wa1, float* __restrict__ wa2,
                            float* __restrict__ wta) {
    int k = threadIdx.x;          // 0..127
    float a1 = 0.f, a2 = 0.f, at = 0.f;
    #pragma unroll 8
    for (int o = 0; o < 64; ++o) {
        float w = W[o * 128 + k];
        a1 += w * a[o];
        a2 += w * a[64 + o];
        at += Wt[o * 128 + k] * a[128 + o];
    }
    wa1[k] = a1; wa2[k] = a2; wta[k] = at;
}

// 3) per-node attention scalars s1[n] = nf[n]·wa1, s2[n] = nf[n]·wa2
__global__ void tgat_node_scalars(const float* __restrict__ nf,
                                  const float* __restrict__ wa1,
                                  const float* __restrict__ wa2,
                                  float* __restrict__ s1, float* __restrict__ s2, int N) {
    __shared__ float sw1[128], sw2[128];
    int t = threadIdx.x;
    if (t < 128) { sw1[t] = wa1[t]; sw2[t] = wa2[t]; }
    __syncthreads();
    int n = blockIdx.x * blockDim.x + t;
    if (n >= N) return;
    const float* r = nf + (size_t)n * 128;
    float acc1 = 0.f, acc2 = 0.f;
    #pragma unroll 8
    for (int k = 0; k < 128; ++k) {
        float x = r[k];
        acc1 += x * sw1[k];
        acc2 += x * sw2[k];
    }
    s1[n] = acc1; s2[n] = acc2;
}

// 4) v = nf @ Wv^T  via V_WMMA_F32_16X16X4_F32.
//    One wave per 16-node tile; 4 accumulators cover the 64 output columns.
//    A (16x4 f32): lanes 0-15 hold M=lane, K={k,k+1}; lanes 16-31 hold M=lane-16, K={k+2,k+3}.
//    B (4x16 f32): lanes 0-15 hold N=lane,  K={k,k+1}; lanes 16-31 hold N=lane-16, K={k+2,k+3}.
//    D (16x16 f32): VGPR r -> M=r (lanes 0-15) / M=r+8 (lanes 16-31), N = lane&15.
__global__ void tgat_gemm_v(const float* __restrict__ nf, const float* __restrict__ Wv,
                            float* __restrict__ v, int N, int tiles) {
    int lane = threadIdx.x & 31;
    int wave = threadIdx.x >> 5;
    int tile = blockIdx.x * 8 + wave;
    if (tile >= tiles) return;               // wave-uniform: EXEC stays all-1s for WMMA
    int nodeBase = tile * 16;
    int half = lane >> 4;                    // 0 or 1
    int l    = lane & 15;
    int row  = nodeBase + l; if (row >= N) row = N - 1;
    const float* arow = nf + (size_t)row * 128 + half * 2;
    const float* b0p  = Wv + (size_t)l * 128 + half * 2;         // N-tile 0
    const float* b1p  = b0p + 16 * 128;                          // N-tile 1
    const float* b2p  = b0p + 32 * 128;                          // N-tile 2
    const float* b3p  = b0p + 48 * 128;                          // N-tile 3

    v8f acc0 = {}, acc1 = {}, acc2 = {}, acc3 = {};
    #pragma unroll 2
    for (int k = 0; k < 128; k += 4) {
        v2f afr = *(const v2f*)(arow + k);
        v2f b0  = *(const v2f*)(b0p + k);
        v2f b1  = *(const v2f*)(b1p + k);
        v2f b2  = *(const v2f*)(b2p + k);
        v2f b3  = *(const v2f*)(b3p + k);
        acc0 = __builtin_amdgcn_wmma_f32_16x16x4_f32(false, afr, false, b0, (short)0, acc0, false, false);
        acc1 = __builtin_amdgcn_wmma_f32_16x16x4_f32(false, afr, false, b1, (short)0, acc1, false, false);
        acc2 = __builtin_amdgcn_wmma_f32_16x16x4_f32(false, afr, false, b2, (short)0, acc2, false, false);
        acc3 = __builtin_amdgcn_wmma_f32_16x16x4_f32(false, afr, false, b3, (short)0, acc3, false, false);
    }
    int mbase = half * 8;
    float* obase = v + (size_t)(nodeBase + mbase) * 64 + l;
    if (nodeBase + 16 <= N) {
        // full tile: unguarded stores (wave-uniform path, no per-row exec juggling)
        #pragma unroll
        for (int r = 0; r < 8; ++r) {
            float* o = obase + (size_t)r * 64;
            o[0]  = acc0[r];
            o[16] = acc1[r];
            o[32] = acc2[r];
            o[48] = acc3[r];
        }
    } else {
        #pragma unroll
        for (int r = 0; r < 8; ++r) {
            if (nodeBase + mbase + r < N) {
                float* o = obase + (size_t)r * 64;
                o[0]  = acc0[r];
                o[16] = acc1[r];
                o[32] = acc2[r];
                o[48] = acc3[r];
            }
        }
    }
}

// 5) per-edge logit + LeakyReLU + segment max
__global__ void tgat_edge_e(const float* __restrict__ td, const int* __restrict__ src,
                            const int* __restrict__ dst,
                            const float* __restrict__ s1, const float* __restrict__ s2,
                            const float* __restrict__ omega, const float* __restrict__ wta,
                            float* __restrict__ ebuf, float* __restrict__ mx, int E) {
    __shared__ float so[64];
    __shared__ float swt[128];
    int t = threadIdx.x;
    if (t < 64)  so[t]  = omega[t];
    if (t < 128) swt[t] = wta[t];
    __syncthreads();
    int i = blockIdx.x * blockDim.x + t;
    if (i >= E) return;
    float tm = td[i];
    int s = src[i], d = dst[i];
    float acc = 0.f;
    #pragma unroll 8
    for (int j = 0; j < 64; ++j) {
        float sn, cs;
        __sincosf(tm * so[j], &sn, &cs);
        acc += sn * swt[2 * j] + cs * swt[2 * j + 1];
    }
    const float scale = 0.08838834764831845f;   // sqrt(1/128)
    float e = s1[s] + s2[d] + acc * scale;
    e = e > 0.f ? e : 0.2f * e;                 // LeakyReLU(0.2)
    ebuf[i] = e;
    atomicMaxF32(mx + d, e);
}

// 6) ex = exp(e - m[dst]); segment sum
__global__ void tgat_edge_exp(const int* __restrict__ dst, float* __restrict__ ebuf,
                              const float* __restrict__ mx, float* __restrict__ den, int E) {
    int i = blockIdx.x * blockDim.x + threadIdx.x;
    if (i >= E) return;
    int d = dst[i];
    float ex = __expf(ebuf[i] - mx[d]);
    ebuf[i] = ex;
    atomicAdd(den + d, ex);
}

// 7) out[dst] += (ex/den[dst]) * v[src]
//    One thread per (edge, 4-component group): float4 gather of v, 4 f32 atomics.
__global__ void tgat_edge_scatter(const int* __restrict__ src, const int* __restrict__ dst,
                                  const float* __restrict__ ebuf, const float* __restrict__ den,
                                  const float* __restrict__ v, float* __restrict__ out,
                                  long long total4) {
    long long idx = (long long)blockIdx.x * blockDim.x + threadIdx.x;
    if (idx >= total4) return;
    int i  = (int)(idx >> 4);          // edge
    int o4 = (int)(idx & 15) * 4;      // component group
    int d = dst[i], s = src[i];
    float alpha = ebuf[i] / den[d];
    float4 vv = *(const float4*)(v + (size_t)s * 64 + o4);
    float* op = out + (size_t)d * 64 + o4;
    atomicAdd(op + 0, alpha * vv.x);
    atomicAdd(op + 1, alpha * vv.y);
    atomicAdd(op + 2, alpha * vv.z);
    atomicAdd(op + 3, alpha * vv.w);
}

extern "C" void kernel_launch(void* const* d_in, const int* in_sizes, int n_in,
                              void* d_out, int out_size, void* d_ws, size_t ws_size,
                              hipStream_t stream) {
    const float* nf    = (const float*)d_in[0];
    const float* td    = (const float*)d_in[1];
    const int*   src   = (const int*)d_in[2];
    const int*   dst   = (const int*)d_in[3];
    const float* W     = (const float*)d_in[4];
    const float* Wv    = (const float*)d_in[5];
    const float* omega = (const float*)d_in[6];
    const float* Wt    = (const float*)d_in[7];
    const float* a     = (const float*)d_in[8];
    float* out = (float*)d_out;

    const int IN = 128, OUT = 64;
    int N = in_sizes[0] / IN;      // 50000
    int E = in_sizes[1];           // 800000

    float* ws   = (float*)d_ws;
    float* wa1  = ws;                      // 128
    float* wa2  = ws + 128;                // 128
    float* wta  = ws + 256;                // 128
    float* s1   = ws + 384;                // N
    float* s2   = s1 + N;                  // N
    float* mx   = s2 + N;                  // N
    float* den  = mx + N;                  // N
    float* ebuf = den + ((N + 3) & ~3);    // E   (keep later buffers 16B-aligned)
    float* v    = ebuf + ((E + 3) & ~3);   // N*64, 16B-aligned for float4 gathers

    int totalInit = N * OUT;
    tgat_init<<<(totalInit + 255) / 256, 256, 0, stream>>>(out, mx, den, N, totalInit);
    tgat_fold_a<<<1, 128, 0, stream>>>(W, Wt, a, wa1, wa2, wta);
    tgat_node_scalars<<<(N + 255) / 256, 256, 0, stream>>>(nf, wa1, wa2, s1, s2, N);
    int tiles = (N + 15) / 16;
    tgat_gemm_v<<<(tiles + 7) / 8, 256, 0, stream>>>(nf, Wv, v, N, tiles);
    tgat_edge_e<<<(E + 255) / 256, 256, 0, stream>>>(td, src, dst, s1, s2, omega, wta, ebuf, mx, E);
    tgat_edge_exp<<<(E + 255) / 256, 256, 0, stream>>>(dst, ebuf, mx, den, E);
    long long total4 = (long long)E * (OUT / 4);
    tgat_edge_scatter<<<(unsigned)((total4 + 255) / 256), 256, 0, stream>>>(src, dst, ebuf, den, v, out, total4);
}